// GINConv_0P_35399120453944
// MI455X (gfx1250) — compile-verified
//
#include <hip/hip_runtime.h>

#define N_NODES 100000
#define N_EDGES 1600000
#define D_IN 64
#define D_HID 256

// LDS row strides (floats), padded to de-conflict the 64 LDS banks between
// the two 16-lane halves of a wave32. Strides keep rows 16B-aligned so the
// async B128 fills stay legal.
#define W1S 264   // 264 % 64 = 8  -> half-waves (rows k, k+2) shifted by 16 banks
#define W2S 80    // 80  % 64 = 16 -> rows k, k+2 shifted by 32 banks
#define H1S 258   // 258 % 64 = 2, 8B-aligned rows for float2 A-loads

typedef __attribute__((ext_vector_type(2))) float v2f;
typedef __attribute__((ext_vector_type(8))) float v8f;

// ---------------------------------------------------------------------------
// Kernel 1: agg = eps * x   (fuses the (1+eps)-scaled self term into the
// aggregation buffer so the MLP stage reads a single tensor)
// ---------------------------------------------------------------------------
__global__ void gin_init_agg(const float* __restrict__ x,
                             const float* __restrict__ eps,
                             float* __restrict__ agg) {
  int i = blockIdx.x * blockDim.x + threadIdx.x;
  const int n4 = N_NODES * D_IN / 4;
  if (i >= n4) return;
  float e = eps[0];
  float4 v = ((const float4*)x)[i];
  ((float4*)agg)[i] = make_float4(e * v.x, e * v.y, e * v.z, e * v.w);
}

// ---------------------------------------------------------------------------
// Kernel 2: edge scatter-add. 16 lanes cover one 64-float node row via
// float4 gathers + native f32 global atomics (x and agg are L2-resident:
// 2 x 25.6 MB inside the 192 MB L2).
// ---------------------------------------------------------------------------
__global__ void gin_scatter(const float* __restrict__ x,
                            const int* __restrict__ ei,
                            float* __restrict__ agg) {
  unsigned long long gid =
      (unsigned long long)blockIdx.x * blockDim.x + threadIdx.x;
  unsigned e = (unsigned)(gid >> 4);
  if (e >= N_EDGES) return;
  int d = ((int)gid & 15) * 4;
  int src = ei[e];             // edge_index[0][e]
  int dst = ei[N_EDGES + e];   // edge_index[1][e]
  float4 v = *(const float4*)(x + (size_t)src * D_IN + d);
  float* a = agg + (size_t)dst * D_IN + d;
  __hip_atomic_fetch_add(a + 0, v.x, __ATOMIC_RELAXED, __HIP_MEMORY_SCOPE_AGENT);
  __hip_atomic_fetch_add(a + 1, v.y, __ATOMIC_RELAXED, __HIP_MEMORY_SCOPE_AGENT);
  __hip_atomic_fetch_add(a + 2, v.z, __ATOMIC_RELAXED, __HIP_MEMORY_SCOPE_AGENT);
  __hip_atomic_fetch_add(a + 3, v.w, __ATOMIC_RELAXED, __HIP_MEMORY_SCOPE_AGENT);
}

// ---------------------------------------------------------------------------
// Kernel 3: fused MLP  out = relu(h @ W1 + b1) @ W2 + b2  via f32 WMMA.
// 8 waves / block, each wave owns a 16-node tile; weights async-staged to LDS.
// ---------------------------------------------------------------------------
__global__ void __launch_bounds__(256, 1)
gin_mlp(const float* __restrict__ h,
        const float* __restrict__ W1g, const float* __restrict__ b1g,
        const float* __restrict__ W2g, const float* __restrict__ b2g,
        float* __restrict__ out) {
  __shared__ float sW1[D_IN * W1S];        // 67,584 B
  __shared__ float sB1[D_HID];             //  1,024 B
  __shared__ float sW2[D_HID * W2S];       // 81,920 B
  __shared__ float sB2[D_IN];              //    256 B
  __shared__ float sH1[8 * 16 * H1S];      // 132,096 B (per-wave 16x256 tiles)

  const int tid = threadIdx.x;

  // --- Async memory->LDS staging of W1/W2 (GLOBAL_LOAD_ASYNC_TO_LDS_B128,
  //     tracked by ASYNCcnt; no VGPR round trip). One 16B chunk per lane-op;
  //     chunks never straddle a padded LDS row.
  {
    unsigned ldsW1 = (unsigned)(size_t)&sW1[0];
    for (int i4 = tid; i4 < (D_IN * D_HID) / 4; i4 += 256) {
      int r = i4 >> 6, c4 = i4 & 63;                 // 64 x 16B chunks per row
      unsigned laddr = ldsW1 + (unsigned)(r * W1S + c4 * 4) * 4u;
      unsigned long long gaddr = (unsigned long long)(size_t)(W1g + i4 * 4);
      asm volatile("global_load_async_to_lds_b128 %0, %1, off"
                   :: "v"(laddr), "v"(gaddr) : "memory");
    }
    unsigned ldsW2 = (unsigned)(size_t)&sW2[0];
    for (int i4 = tid; i4 < (D_HID * D_IN) / 4; i4 += 256) {
      int r = i4 >> 4, c4 = i4 & 15;                 // 16 x 16B chunks per row
      unsigned laddr = ldsW2 + (unsigned)(r * W2S + c4 * 4) * 4u;
      unsigned long long gaddr = (unsigned long long)(size_t)(W2g + i4 * 4);
      asm volatile("global_load_async_to_lds_b128 %0, %1, off"
                   :: "v"(laddr), "v"(gaddr) : "memory");
    }
  }
  if (tid < D_HID) sB1[tid] = b1g[tid];
  if (tid < D_IN)  sB2[tid] = b2g[tid];
  asm volatile("s_wait_asynccnt 0x0" ::: "memory");
  __syncthreads();

  const int wave = tid >> 5;
  const int lane = tid & 31;
  const int row  = lane & 15;   // M index (A layout) / N index (B,C layouts)
  const int o    = lane >> 4;   // half-wave selector (K split for A/B)
  const int m0   = (blockIdx.x * 8 + wave) * 16;

  // ---- Load A tile (16x64 of h) directly in WMMA f32 A-layout -------------
  // Lane holds row (lane&15); VGPR pair k holds K columns (4k+2o, 4k+2o+1).
  v2f a[16];
  {
    int nr = m0 + row;
    if (nr > N_NODES - 1) nr = N_NODES - 1;   // clamp: EXEC must stay all-1s
    const float* hr = h + (size_t)nr * D_IN + 2 * o;
#pragma unroll
    for (int k = 0; k < 16; ++k)
      a[k] = *(const v2f*)(hr + 4 * k);
  }

  float* mySH1 = sH1 + wave * (16 * H1S);

  // ---- GEMM1: (16x64) @ (64x256) + b1, ReLU, stage to LDS -----------------
  for (int n = 0; n < 16; ++n) {
    float bias = sB1[n * 16 + row];
    v8f acc = {bias, bias, bias, bias, bias, bias, bias, bias};
    const float* bcol = sW1 + (2 * o) * W1S + n * 16 + row;
#pragma unroll
    for (int k = 0; k < 16; ++k) {
      v2f bv;
      bv.x = bcol[(4 * k) * W1S];
      bv.y = bcol[(4 * k + 1) * W1S];
      acc = __builtin_amdgcn_wmma_f32_16x16x4_f32(
          false, a[k], false, bv, (short)0, acc, false, false);
    }
    // ReLU + C-layout -> row-major LDS (per-wave region, in-order DS ops)
#pragma unroll
    for (int r = 0; r < 8; ++r) {
      mySH1[(r + 8 * o) * H1S + n * 16 + row] = __builtin_fmaxf(acc[r], 0.0f);
    }
  }

  // ---- GEMM2: (16x256) @ (256x64) + b2 ------------------------------------
  const float* arow = mySH1 + row * H1S + 2 * o;
  for (int n = 0; n < 4; ++n) {
    float bias = sB2[n * 16 + row];
    v8f acc = {bias, bias, bias, bias, bias, bias, bias, bias};
    const float* bcol = sW2 + (2 * o) * W2S + n * 16 + row;
#pragma unroll 8
    for (int k = 0; k < 64; ++k) {
      v2f av = *(const v2f*)(arow + 4 * k);
      v2f bv;
      bv.x = bcol[(4 * k) * W2S];
      bv.y = bcol[(4 * k + 1) * W2S];
      acc = __builtin_amdgcn_wmma_f32_16x16x4_f32(
          false, av, false, bv, (short)0, acc, false, false);
    }
#pragma unroll
    for (int r = 0; r < 8; ++r) {
      int node = m0 + r + 8 * o;
      if (node < N_NODES)
        out[(size_t)node * D_IN + n * 16 + row] = acc[r];
    }
  }
}

// ---------------------------------------------------------------------------
extern "C" void kernel_launch(void* const* d_in, const int* in_sizes, int n_in,
                              void* d_out, int out_size, void* d_ws,
                              size_t ws_size, hipStream_t stream) {
  const float* x   = (const float*)d_in[0];
  const float* eps = (const float*)d_in[1];
  const float* W1  = (const float*)d_in[2];
  const float* b1  = (const float*)d_in[3];
  const float* W2  = (const float*)d_in[4];
  const float* b2  = (const float*)d_in[5];
  const int*   ei  = (const int*)d_in[6];
  float* out = (float*)d_out;
  float* agg = (float*)d_ws;   // N_NODES * D_IN floats = 25.6 MB scratch

  const int n4 = N_NODES * D_IN / 4;
  gin_init_agg<<<(n4 + 255) / 256, 256, 0, stream>>>(x, eps, agg);

  const long long sthreads = (long long)N_EDGES * 16;
  gin_scatter<<<(int)((sthreads + 255) / 256), 256, 0, stream>>>(x, ei, agg);

  gin_mlp<<<(N_NODES + 127) / 128, 256, 0, stream>>>(agg, W1, b1, W2, b2, out);
}